// InterpolatedCDTriMesh_9758165696609
// MI455X (gfx1250) — compile-verified
//
#include <hip/hip_runtime.h>
#include <math.h>

// ---------------------------------------------------------------------------
// InterpolatedCDTriMesh chamfer-style loss for MI455X (gfx1250, wave32, WMMA)
// Core pairwise term  -2 * pts . shape  computed with V_WMMA_F32_16X16X4_F32
// ---------------------------------------------------------------------------

#define B_  4
#define M_  642
#define F_  800
#define N_  8192
#define K_  8              // fp32 linspace numerics exclude (1/3,2/3) & (2/3,1/3)
#define P_  (F_ * K_)      // 6400 interpolated points per batch
#define PT_ (P_ / 16)      // 400 p-tiles per batch
#define NT_ (N_ / 16)      // 512 n-tiles per batch
#define CHUNK_  4          // p-tiles per wave
#define CHUNKS_ (PT_ / CHUNK_)   // 100 chunks per batch
#define HS_LAMBDA_ 0.05f

typedef __attribute__((ext_vector_type(2))) float v2f;
typedef __attribute__((ext_vector_type(8))) float v8f;

// Exact fp32 barycentric weights surviving the (1.0f - g0) - g1 >= 0 filter,
// in meshgrid('ij') row-major order. Bit-exact hexfloat constants.
__constant__ float WG0[K_] = {0.0f, 0.0f, 0.0f, 0.0f,
                              0x1.555556p-2f, 0x1.555556p-2f, 0x1.555556p-1f, 1.0f};
__constant__ float WG1[K_] = {0.0f, 0x1.555556p-2f, 0x1.555556p-1f, 1.0f,
                              0.0f, 0x1.555556p-2f, 0.0f, 0.0f};
__constant__ float WG2[K_] = {1.0f, 0x1.555554p-1f, 0x1.555554p-2f, 0.0f,
                              0x1.555554p-1f, 0x1.555552p-2f, 0x1.555554p-2f, 0.0f};

// Order-preserving float <-> uint map so unsigned atomicMin == float min.
__device__ __forceinline__ unsigned encf(float x) {
    unsigned u = __float_as_uint(x);
    return (u & 0x80000000u) ? ~u : (u | 0x80000000u);
}
__device__ __forceinline__ float decf(unsigned e) {
    return (e & 0x80000000u) ? __uint_as_float(e & 0x7FFFFFFFu)
                             : __uint_as_float(~e);
}
__device__ __forceinline__ float hshrink(float x) {
    return (fabsf(x) > HS_LAMBDA_) ? x : 0.0f;
}

// ---------------------------------------------------------------------------
// K0: init shape->cage mins to enc(+inf), zero the scalar output
// ---------------------------------------------------------------------------
__global__ void k0_init(unsigned* __restrict__ s2c_enc, float* __restrict__ out) {
    int idx = blockIdx.x * blockDim.x + threadIdx.x;
    if (idx < B_ * N_) s2c_enc[idx] = 0xFF800000u;   // enc(+inf)
    if (idx == 0) out[0] = 0.0f;
}

// ---------------------------------------------------------------------------
// K1a: gather face vertices, barycentric-interpolate -> pts4 (xyz0) + |p|^2
// ---------------------------------------------------------------------------
__global__ void k1_pts(const float* __restrict__ cage_v,
                       const long long* __restrict__ cage_f,
                       float* __restrict__ pts4, float* __restrict__ pp) {
    int idx = blockIdx.x * blockDim.x + threadIdx.x;
    if (idx >= B_ * F_ * K_) return;
    int k = idx % K_;
    int f = (idx / K_) % F_;
    int b = idx / (K_ * F_);

    const long long* fi = cage_f + ((size_t)b * F_ + f) * 3;
    const float* v0 = cage_v + ((size_t)b * M_ + (int)fi[0]) * 3;
    const float* v1 = cage_v + ((size_t)b * M_ + (int)fi[1]) * 3;
    const float* v2 = cage_v + ((size_t)b * M_ + (int)fi[2]) * 3;

    float w0 = WG0[k], w1 = WG1[k], w2 = WG2[k];
    float x = w0 * v0[0] + w1 * v1[0] + w2 * v2[0];
    float y = w0 * v0[1] + w1 * v1[1] + w2 * v2[1];
    float z = w0 * v0[2] + w1 * v1[2] + w2 * v2[2];

    size_t o = (size_t)b * P_ + (size_t)f * K_ + k;
    pts4[o * 4 + 0] = x; pts4[o * 4 + 1] = y;
    pts4[o * 4 + 2] = z; pts4[o * 4 + 3] = 0.0f;
    pp[o] = x * x + y * y + z * z;
}

// ---------------------------------------------------------------------------
// K1b: pad shape to xyz0 + |s|^2
// ---------------------------------------------------------------------------
__global__ void k1_shape(const float* __restrict__ shape,
                         float* __restrict__ shp4, float* __restrict__ ss) {
    int idx = blockIdx.x * blockDim.x + threadIdx.x;
    if (idx >= B_ * N_) return;
    const float* s = shape + (size_t)idx * 3;
    float x = s[0], y = s[1], z = s[2];
    shp4[idx * 4 + 0] = x; shp4[idx * 4 + 1] = y;
    shp4[idx * 4 + 2] = z; shp4[idx * 4 + 3] = 0.0f;
    ss[idx] = x * x + y * y + z * z;
}

// ---------------------------------------------------------------------------
// K2: WMMA pairwise distances + row/col mins.
// One wave owns CHUNK_ p-tiles of one batch, sweeps all n-tiles.
// A frag (16x4 f32): lane L<16 -> row L, K={0,1}; lane L>=16 -> row L-16, K={2,3}
// B frag (4x16 f32): mirrored (lane -> col, half-wave selects K pair)
// C/D (16x16 f32, 8 VGPRs): lane -> col (lane&15), VGPR r -> row r + 8*(lane>>4)
// ---------------------------------------------------------------------------
__global__ void __launch_bounds__(256)
k2_dist(const float* __restrict__ pts4, const float* __restrict__ pp,
        const float* __restrict__ shp4, const float* __restrict__ ss,
        float* __restrict__ c2s, unsigned* __restrict__ s2c_enc) {
    const int b     = blockIdx.y;
    const int chunk = blockIdx.x * 8 + (threadIdx.x >> 5);
    if (chunk >= CHUNKS_) return;                 // wave-uniform exit
    const int lane = threadIdx.x & 31;
    const int lo   = lane & 15;
    const int hi   = lane >> 4;
    const int prow0 = chunk * (CHUNK_ * 16);
    const float INF = __builtin_inff();

    v2f   a[CHUNK_];
    float pm[CHUNK_][8];      // |p|^2 for the 8 C rows of each tile
    float racc[CHUNK_][8];    // running min over n of (|s|^2 - 2 p.s)

    #pragma unroll
    for (int j = 0; j < CHUNK_; ++j) {
        int p = prow0 + j * 16 + lo;
        const float* ap = pts4 + (((size_t)b * P_ + p) << 2) + (hi << 1);
        a[j] = (v2f){ap[0], ap[1]};
        #pragma unroll
        for (int r = 0; r < 8; ++r) {
            pm[j][r]   = pp[(size_t)b * P_ + prow0 + j * 16 + r + 8 * hi];
            racc[j][r] = INF;
        }
    }

    for (int nt = 0; nt < NT_; ++nt) {
        int n = nt * 16 + lo;
        const float* bp = shp4 + (((size_t)b * N_ + n) << 2) + (hi << 1);
        v2f   bm  = (v2f){bp[0], bp[1]};
        float ssv = ss[(size_t)b * N_ + n];
        float cm  = INF;                          // min over the chunk's 64 rows of (|p|^2 - 2 p.s)

        #pragma unroll
        for (int j = 0; j < CHUNK_; ++j) {
            v8f c = {};
            // D = A(16x4) x B(4x16): dot(p, s) per (row,col), K padded with 0
            c = __builtin_amdgcn_wmma_f32_16x16x4_f32(
                    false, a[j], false, bm, (short)0, c, false, false);
            #pragma unroll
            for (int r = 0; r < 8; ++r) {
                float m2c = -2.0f * c[r];
                racc[j][r] = fminf(racc[j][r], ssv + m2c);
                cm         = fminf(cm, pm[j][r] + m2c);
            }
        }
        cm = fminf(cm, __shfl_xor(cm, 16, 32));   // combine both row halves
        if (hi == 0)
            atomicMin(&s2c_enc[(size_t)b * N_ + n], encf(ssv + cm));
    }

    // row mins: reduce across the 16 columns held by lanes of each half-wave
    #pragma unroll
    for (int j = 0; j < CHUNK_; ++j) {
        #pragma unroll
        for (int r = 0; r < 8; ++r) {
            float v = racc[j][r];
            v = fminf(v, __shfl_xor(v, 1, 32));
            v = fminf(v, __shfl_xor(v, 2, 32));
            v = fminf(v, __shfl_xor(v, 4, 32));
            v = fminf(v, __shfl_xor(v, 8, 32));
            if (lo == 0)
                c2s[(size_t)b * P_ + prow0 + j * 16 + r + 8 * hi] = pm[j][r] + v;
        }
    }
}

// ---------------------------------------------------------------------------
// K3: per-batch hardshrink + mean/mean/max, atomicAdd loss/B into out[0]
// ---------------------------------------------------------------------------
__global__ void k3_reduce(const float* __restrict__ c2s,
                          const unsigned* __restrict__ s2c_enc,
                          float* __restrict__ out) {
    const int b = blockIdx.x, tid = threadIdx.x;
    float ssum = 0.0f, csum = 0.0f, cmax = -__builtin_inff();
    for (int n = tid; n < N_; n += 256)
        ssum += hshrink(decf(s2c_enc[(size_t)b * N_ + n]));
    for (int p = tid; p < P_; p += 256) {
        float v = hshrink(c2s[(size_t)b * P_ + p]);
        csum += v;
        cmax  = fmaxf(cmax, v);
    }
    __shared__ float sh[256];
    sh[tid] = ssum; __syncthreads();
    for (int s = 128; s > 0; s >>= 1) { if (tid < s) sh[tid] += sh[tid + s]; __syncthreads(); }
    float tot_s = sh[0]; __syncthreads();
    sh[tid] = csum; __syncthreads();
    for (int s = 128; s > 0; s >>= 1) { if (tid < s) sh[tid] += sh[tid + s]; __syncthreads(); }
    float tot_c = sh[0]; __syncthreads();
    sh[tid] = cmax; __syncthreads();
    for (int s = 128; s > 0; s >>= 1) { if (tid < s) sh[tid] = fmaxf(sh[tid], sh[tid + s]); __syncthreads(); }
    if (tid == 0) {
        float loss_b = tot_s * (1.0f / N_)   // mean(s2c) * (GAMMA + DELTA*M) = *1
                     + tot_c * (1.0f / P_)   // mean(c2s)
                     + sh[0];                // BETA * max(c2s) = *1
        atomicAdd(out, loss_b * (1.0f / B_));
    }
}

// ---------------------------------------------------------------------------
// Workspace layout (bytes, 16B aligned)
// ---------------------------------------------------------------------------
#define WS_PTS4  0                                   // B*P*4 f32 = 409600
#define WS_PP    (WS_PTS4 + B_ * P_ * 4 * 4)         // B*P   f32 = 102400
#define WS_SHP4  (WS_PP   + B_ * P_ * 4)             // B*N*4 f32 = 524288
#define WS_SS    (WS_SHP4 + B_ * N_ * 4 * 4)         // B*N   f32 = 131072
#define WS_C2S   (WS_SS   + B_ * N_ * 4)             // B*P   f32 = 102400
#define WS_S2C   (WS_C2S  + B_ * P_ * 4)             // B*N   u32 = 131072

extern "C" void kernel_launch(void* const* d_in, const int* in_sizes, int n_in,
                              void* d_out, int out_size, void* d_ws, size_t ws_size,
                              hipStream_t stream) {
    const float*     cage_v = (const float*)d_in[0];
    const long long* cage_f = (const long long*)d_in[1];   // int64 indices
    const float*     shape  = (const float*)d_in[2];
    float* out = (float*)d_out;

    char* ws = (char*)d_ws;
    float*    pts4 = (float*)(ws + WS_PTS4);
    float*    pp   = (float*)(ws + WS_PP);
    float*    shp4 = (float*)(ws + WS_SHP4);
    float*    ss   = (float*)(ws + WS_SS);
    float*    c2s  = (float*)(ws + WS_C2S);
    unsigned* s2c  = (unsigned*)(ws + WS_S2C);

    k0_init <<<(B_ * N_ + 255) / 256, 256, 0, stream>>>(s2c, out);
    k1_pts  <<<(B_ * F_ * K_ + 255) / 256, 256, 0, stream>>>(cage_v, cage_f, pts4, pp);
    k1_shape<<<(B_ * N_ + 255) / 256, 256, 0, stream>>>(shape, shp4, ss);

    dim3 g2((CHUNKS_ + 7) / 8, B_);
    k2_dist <<<g2, 256, 0, stream>>>(pts4, pp, shp4, ss, c2s, s2c);

    k3_reduce<<<B_, 256, 0, stream>>>(c2s, s2c, out);
}